// DeformableSpatialAttentionLayer_9156870275466
// MI455X (gfx1250) — compile-verified
//
#include <hip/hip_runtime.h>

#define EMBED 256
#define HEADS 8
#define PTS 12
#define DPH 32          // dim_per_head (value proj width)
#define HEAD_D 4
#define BSZ 8
#define HGT 128
#define WID 128
#define NQ (HGT * WID)          // 16384
#define RTOT (BSZ * NQ)         // 131072 rows
#define NOFF (HEADS * PTS * 2)  // 192
#define NATT (HEADS * PTS)      // 96

typedef float v2f __attribute__((ext_vector_type(2)));
typedef float v8f __attribute__((ext_vector_type(8)));

// ---------------------------------------------------------------------------
// Per-wave GEMM helper: 16(M) x 48(N = 3 tiles) x 256(K) with f32 WMMA.
// LDB is compile-time so B/output addressing stays 32-bit int offsets off the
// kernel-arg pointer (lowers to global_load saddr+voffset scale_offset form).
// ---------------------------------------------------------------------------
template<int LDB>
__device__ __forceinline__ void qproj_wave_gemm(
    const float* Alds,                   // per-lane A fragment base (LDS)
    const float* __restrict__ W,         // (256, LDB) row-major weights
    const float* __restrict__ bias,      // (LDB) bias
    float* __restrict__ outp,            // (RTOT, LDB) output
    int c0,                              // first column = tile0*16 + nsel
    int khalf, int rbase)
{
    v8f acc[3] = {v8f{}, v8f{}, v8f{}};
    for (int k = 0; k < EMBED; k += 4) {
        v2f a = *(const v2f*)(Alds + k);
        int kr = k + khalf;
        #pragma unroll
        for (int i = 0; i < 3; ++i) {
            int c = c0 + i * 16;
            v2f b;
            b.x = W[kr * LDB + c];
            b.y = W[(kr + 1) * LDB + c];
            acc[i] = __builtin_amdgcn_wmma_f32_16x16x4_f32(
                false, a, false, b, (short)0, acc[i], false, false);
        }
    }
    #pragma unroll
    for (int i = 0; i < 3; ++i) {
        int c = c0 + i * 16;
        float bi = bias[c];
        #pragma unroll
        for (int rr = 0; rr < 8; ++rr)
            outp[(size_t)(rbase + rr) * LDB + c] = acc[i][rr] + bi;
    }
}

// ---------------------------------------------------------------------------
// Kernel 1: off = query @ W_off + b_off ; attn_raw = query @ W_attn + b_attn
// Block: 192 threads = 6 waves. Waves 0-3 -> 12 W_off tiles (3 each),
// waves 4-5 -> 6 W_attn tiles (3 each). Uniform branch, no divergence.
// 16-row query M-tile staged once in LDS (padded stride 260).
// ---------------------------------------------------------------------------
__global__ __launch_bounds__(192) void dsal_qproj(
    const float* __restrict__ query,
    const float* __restrict__ W_off, const float* __restrict__ b_off,
    const float* __restrict__ W_attn, const float* __restrict__ b_attn,
    float* __restrict__ off_out, float* __restrict__ attn_out)
{
    __shared__ float Atile[16 * 260];   // padded stride 260 -> conflict-free
    const int tid  = threadIdx.x;
    const int m0   = blockIdx.x << 4;
    const int wave = tid >> 5;
    const int lane = tid & 31;
    const int nsel = lane & 15;
    const int khalf = (lane >> 4) << 1;     // 0 or 2

    for (int idx = tid; idx < 16 * EMBED; idx += 192) {
        int r = idx >> 8, kk = idx & 255;
        Atile[r * 260 + kk] = query[(size_t)(m0 + r) * EMBED + kk];
    }
    __syncthreads();

    const float* Alds = Atile + nsel * 260 + khalf;
    const int rbase = m0 + ((lane >> 4) << 3);

    if (wave < 4) {
        // W_off: tiles {3w, 3w+1, 3w+2} of 12 -> columns [w*48, w*48+47]
        qproj_wave_gemm<NOFF>(Alds, W_off, b_off, off_out,
                              wave * 48 + nsel, khalf, rbase);
    } else {
        // W_attn: tiles {3(w-4) ...} of 6 -> columns [(w-4)*48, ...+47]
        qproj_wave_gemm<NATT>(Alds, W_attn, b_attn, attn_out,
                              (wave - 4) * 48 + nsel, khalf, rbase);
    }
}

// ---------------------------------------------------------------------------
// Kernel 2: v = value @ W_val + b_val, scattered to ws as (b, head, hw, 4)
// Block: 128 threads = 4 waves = 2 M-subtiles x 2 N-tiles (32 rows / block)
// ---------------------------------------------------------------------------
__global__ __launch_bounds__(128) void dsal_vproj(
    const float* __restrict__ value,
    const float* __restrict__ W_val, const float* __restrict__ b_val,
    float* __restrict__ v_ws)
{
    __shared__ float Atile[32 * 260];
    const int tid  = threadIdx.x;
    const int mblk = blockIdx.x << 5;
    const int wave = tid >> 5;
    const int lane = tid & 31;
    const int nsel = lane & 15;
    const int khalf = (lane >> 4) << 1;

    for (int idx = tid; idx < 32 * EMBED; idx += 128) {
        int r = idx >> 8, kk = idx & 255;
        Atile[r * 260 + kk] = value[(size_t)(mblk + r) * EMBED + kk];
    }
    __syncthreads();

    const int msub = wave >> 1;             // 0/1
    const int n0   = (wave & 1) << 4;       // 0/16
    const int m0   = mblk + (msub << 4);
    const int c    = n0 + nsel;             // 0..31

    v8f acc = {};
    const float* Alds = Atile + (msub << 4) * 260 + nsel * 260 + khalf;
    for (int k = 0; k < EMBED; k += 4) {
        v2f a = *(const v2f*)(Alds + k);
        int kr = k + khalf;
        v2f b;
        b.x = W_val[kr * DPH + c];
        b.y = W_val[(kr + 1) * DPH + c];
        acc = __builtin_amdgcn_wmma_f32_16x16x4_f32(
            false, a, false, b, (short)0, acc, false, false);
    }

    const int h  = c >> 2, dd = c & 3;
    const float bias = b_val[c];
    const int rbase = m0 + ((lane >> 4) << 3);
    #pragma unroll
    for (int rr = 0; rr < 8; ++rr) {
        int row = rbase + rr;
        int b_ = row >> 14;                 // / NQ
        int q  = row & (NQ - 1);
        v_ws[((size_t)(b_ * HEADS + h) * NQ + q) * HEAD_D + dd] = acc[rr] + bias;
    }
}

// ---------------------------------------------------------------------------
// Kernel 3: softmax (written back as the attn output) + bilinear gather into
// LDS agg, then 16x32x256 output GEMM (f32 WMMA) + bias + identity.
// Block: 128 threads handles 16 query rows. Gather thread = (q_local, head).
// ---------------------------------------------------------------------------
#define LDSW 36   // padded agg stride (conflict-free, 8B-aligned fragments)

__global__ __launch_bounds__(128) void dsal_gather_out(
    const float* __restrict__ query,
    const float* __restrict__ W_out, const float* __restrict__ b_out,
    const float* __restrict__ v_ws,
    const float* __restrict__ off_in,
    float* __restrict__ attn_io,
    float* __restrict__ out)
{
    __shared__ float lds_agg[16 * LDSW];
    const int tid = threadIdx.x;
    const int m0  = blockIdx.x << 4;

    {   // ---- gather phase ----
        const int ql  = tid >> 3;
        const int h   = tid & 7;
        const int row = m0 + ql;
        const int b_  = row >> 14;
        const int q   = row & (NQ - 1);
        const int iy  = q >> 7;
        const int ix  = q & (WID - 1);

        float logits[PTS];
        const size_t abase = (size_t)row * NATT + h * PTS;
        float mx = -1e30f;
        #pragma unroll
        for (int p = 0; p < PTS; ++p) {
            logits[p] = attn_io[abase + p];
            mx = fmaxf(mx, logits[p]);
        }
        float s = 0.f;
        #pragma unroll
        for (int p = 0; p < PTS; ++p) { logits[p] = __expf(logits[p] - mx); s += logits[p]; }
        const float inv = 1.0f / s;

        const float* vhb = v_ws + (size_t)(b_ * HEADS + h) * NQ * HEAD_D;
        const size_t obase = (size_t)row * NOFF + h * (PTS * 2);
        float a0 = 0.f, a1 = 0.f, a2 = 0.f, a3 = 0.f;
        #pragma unroll
        for (int p = 0; p < PTS; ++p) {
            float wgt = logits[p] * inv;
            attn_io[abase + p] = wgt;                 // softmaxed attn output
            float x = (float)ix + off_in[obase + p * 2 + 0];
            float y = (float)iy + off_in[obase + p * 2 + 1];
            float x0f = floorf(x), y0f = floorf(y);
            int   x0 = (int)x0f,  y0 = (int)y0f;
            float wx = x - x0f,   wy = y - y0f;
            #pragma unroll
            for (int cy = 0; cy < 2; ++cy) {
                int yc = y0 + cy;
                float wyc = cy ? wy : (1.f - wy);
                if ((unsigned)yc < (unsigned)HGT) {
                    #pragma unroll
                    for (int cx = 0; cx < 2; ++cx) {
                        int xc = x0 + cx;
                        float cw = wgt * wyc * (cx ? wx : (1.f - wx));
                        if ((unsigned)xc < (unsigned)WID) {
                            const float* src = vhb + (size_t)(yc * WID + xc) * HEAD_D;
                            a0 += cw * src[0]; a1 += cw * src[1];
                            a2 += cw * src[2]; a3 += cw * src[3];
                        }
                    }
                }
            }
        }
        lds_agg[ql * LDSW + h * 4 + 0] = a0;
        lds_agg[ql * LDSW + h * 4 + 1] = a1;
        lds_agg[ql * LDSW + h * 4 + 2] = a2;
        lds_agg[ql * LDSW + h * 4 + 3] = a3;
    }
    __syncthreads();

    // ---- output projection: 4 waves x 4 N-tiles, K = 32 ----
    const int wave = tid >> 5;
    const int lane = tid & 31;
    const int nsel = lane & 15;
    const int khalf = (lane >> 4) << 1;

    v8f acc[4] = {v8f{}, v8f{}, v8f{}, v8f{}};
    const float* Alds = lds_agg + nsel * LDSW + khalf;
    for (int k = 0; k < DPH; k += 4) {
        v2f a = *(const v2f*)(Alds + k);
        int kr = k + khalf;
        #pragma unroll
        for (int i = 0; i < 4; ++i) {
            int c = (((wave << 2) + i) << 4) + nsel;
            v2f b;
            b.x = W_out[kr * EMBED + c];
            b.y = W_out[(kr + 1) * EMBED + c];
            acc[i] = __builtin_amdgcn_wmma_f32_16x16x4_f32(
                false, a, false, b, (short)0, acc[i], false, false);
        }
    }

    const int rbase = m0 + ((lane >> 4) << 3);
    #pragma unroll
    for (int i = 0; i < 4; ++i) {
        int c = (((wave << 2) + i) << 4) + nsel;
        float bias = b_out[c];
        #pragma unroll
        for (int rr = 0; rr < 8; ++rr) {
            int row = rbase + rr;
            out[(size_t)row * EMBED + c] =
                acc[i][rr] + bias + query[(size_t)row * EMBED + c];
        }
    }
}

// ---------------------------------------------------------------------------
extern "C" void kernel_launch(void* const* d_in, const int* in_sizes, int n_in,
                              void* d_out, int out_size, void* d_ws, size_t ws_size,
                              hipStream_t stream) {
    (void)in_sizes; (void)n_in; (void)out_size; (void)ws_size;
    const float* query  = (const float*)d_in[0];
    const float* value  = (const float*)d_in[1];
    const float* W_off  = (const float*)d_in[2];
    const float* b_off  = (const float*)d_in[3];
    const float* W_attn = (const float*)d_in[4];
    const float* b_attn = (const float*)d_in[5];
    const float* W_val  = (const float*)d_in[6];
    const float* b_val  = (const float*)d_in[7];
    const float* W_out  = (const float*)d_in[8];
    const float* b_out  = (const float*)d_in[9];

    float* out  = (float*)d_out;                      // (bs, nq, 256)
    float* offo = out  + (size_t)RTOT * EMBED;        // (bs, nq, 8, 12, 2)
    float* atto = offo + (size_t)RTOT * NOFF;         // (bs, nq, 8, 12)
    float* v_ws = (float*)d_ws;                       // (bs, 8, nq, 4) = 16 MB

    dsal_qproj<<<RTOT / 16, 192, 0, stream>>>(query, W_off, b_off, W_attn, b_attn, offo, atto);
    dsal_vproj<<<RTOT / 32, 128, 0, stream>>>(value, W_val, b_val, v_ws);
    dsal_gather_out<<<RTOT / 16, 128, 0, stream>>>(query, W_out, b_out, v_ws, offo, atto, out);
}